// SE3GeodesicLoss_63359357551120
// MI455X (gfx1250) — compile-verified
//
#include <hip/hip_runtime.h>

typedef __attribute__((ext_vector_type(2))) float v2f;
typedef __attribute__((ext_vector_type(8))) float v8f;

#define EPS_T2 1e-8f

__device__ __forceinline__ void cross3(const float a[3], const float b[3], float o[3]) {
  o[0] = a[1]*b[2] - a[2]*b[1];
  o[1] = a[2]*b[0] - a[0]*b[2];
  o[2] = a[0]*b[1] - a[1]*b[0];
}

// phi = so3_log(R), R row-major 3x3
__device__ __forceinline__ void so3_log_f(const float R[9], float phi[3]) {
  float tr = R[0] + R[4] + R[8];
  float c  = 0.5f * (tr - 1.0f);
  c = fminf(fmaxf(c, -1.0f + 1e-7f), 1.0f - 1e-7f);
  float th = acosf(c);
  float factor = th / (2.0f * sinf(th));
  phi[0] = factor * (R[7] - R[5]);
  phi[1] = factor * (R[2] - R[6]);
  phi[2] = factor * (R[3] - R[1]);
}

// o = so3_inv_left_jac(phi) @ v   (J^-1 = I - 0.5 K + D K^2, K v = phi x v)
__device__ __forceinline__ void invJ_apply(const float phi[3], const float v[3], float o[3]) {
  float t2 = phi[0]*phi[0] + phi[1]*phi[1] + phi[2]*phi[2];
  bool  sm = t2 < EPS_T2;
  float t2s = sm ? 1.0f : t2;
  float th  = sqrtf(t2s);
  float D   = sm ? (1.0f/12.0f + t2/720.0f)
                 : (1.0f/t2s - (1.0f + cosf(th)) / (2.0f * th * sinf(th)));
  float Kv[3], KKv[3];
  cross3(phi, v,  Kv);
  cross3(phi, Kv, KKv);
  #pragma unroll
  for (int i = 0; i < 3; ++i) o[i] = v[i] - 0.5f*Kv[i] + D*KKv[i];
}

// R,t = se3_exp([rho, phi]) top 3x4 block
__device__ __forceinline__ void se3_exp_f(const float rho[3], const float phi[3],
                                          float R[9], float t[3]) {
  float t2 = phi[0]*phi[0] + phi[1]*phi[1] + phi[2]*phi[2];
  bool  sm = t2 < EPS_T2;
  float t2s = sm ? 1.0f : t2;
  float th  = sqrtf(t2s);
  float sA = sm ? (1.0f - t2/6.0f)        : (sinf(th)/th);
  float sB = sm ? (0.5f - t2/24.0f)       : ((1.0f - cosf(th))/t2s);
  float sC = sm ? (1.0f/6.0f - t2/120.0f) : ((th - sinf(th))/(t2s*th));
  float x = phi[0], y = phi[1], z = phi[2];
  // R = I + sA*K + sB*K^2 with K^2 = phi phi^T - t2 I
  R[0] = 1.0f + sB*(x*x - t2); R[1] = -sA*z + sB*x*y;       R[2] =  sA*y + sB*x*z;
  R[3] =  sA*z + sB*x*y;       R[4] = 1.0f + sB*(y*y - t2); R[5] = -sA*x + sB*y*z;
  R[6] = -sA*y + sB*x*z;       R[7] =  sA*x + sB*y*z;       R[8] = 1.0f + sB*(z*z - t2);
  float Kr[3], KKr[3];
  cross3(phi, rho, Kr);
  cross3(phi, Kr,  KKr);
  #pragma unroll
  for (int i = 0; i < 3; ++i) t[i] = rho[i] + sB*Kr[i] + sC*KKr[i];
}

__global__ __launch_bounds__(256) void se3_loss_main(
    const float* __restrict__ input,      // (B,6)
    const float* __restrict__ target,     // (B,4,4) row-major
    const float* __restrict__ precision,  // (6,6)
    float* __restrict__ partials,         // (nblocks,)
    int B)
{
  __shared__ float lds[512 * 16];  // rows 0..255: g (16-padded), 256..511: xi*
  __shared__ float red[256];

  const int tid = threadIdx.x;
  const long long b = (long long)blockIdx.x * 256 + tid;

  float g[6], xs[6];
  if (b < (long long)B) {
    const float* xi = input  + b * 6;
    const float* T  = target + b * 16;
    float RT[9], tT[3];
    #pragma unroll
    for (int r = 0; r < 3; ++r) {
      #pragma unroll
      for (int c = 0; c < 3; ++c) RT[r*3 + c] = T[r*4 + c];
      tT[r] = T[r*4 + 3];
    }
    // xi* = se3_log(target_T_inv)
    float phs[3], rhs_[3];
    so3_log_f(RT, phs);
    invJ_apply(phs, tT, rhs_);
    xs[0] = rhs_[0]; xs[1] = rhs_[1]; xs[2] = rhs_[2];
    xs[3] = phs[0];  xs[4] = phs[1];  xs[5] = phs[2];

    // exp(input)
    float rho[3] = {xi[0], xi[1], xi[2]};
    float phi[3] = {xi[3], xi[4], xi[5]};
    float Re[9], te[3];
    se3_exp_f(rho, phi, Re, te);

    // G = exp(input) @ target_T_inv (top 3x4)
    float Rg[9], tg[3];
    #pragma unroll
    for (int r = 0; r < 3; ++r) {
      #pragma unroll
      for (int c = 0; c < 3; ++c)
        Rg[r*3 + c] = Re[r*3+0]*RT[0*3+c] + Re[r*3+1]*RT[1*3+c] + Re[r*3+2]*RT[2*3+c];
      tg[r] = Re[r*3+0]*tT[0] + Re[r*3+1]*tT[1] + Re[r*3+2]*tT[2] + te[r];
    }
    // g = se3_log(G)
    float phg[3], rhg[3];
    so3_log_f(Rg, phg);
    invJ_apply(phg, tg, rhg);
    g[0] = rhg[0]; g[1] = rhg[1]; g[2] = rhg[2];
    g[3] = phg[0]; g[4] = phg[1]; g[5] = phg[2];
  } else {
    #pragma unroll
    for (int j = 0; j < 6; ++j) { g[j] = 0.0f; xs[j] = 0.0f; }
  }

  // Stage 16-padded rows into LDS (zeros for cols 6..15 keep the Gram exact)
  #pragma unroll
  for (int j = 0; j < 16; ++j) {
    lds[tid * 16 + j]         = (j < 6) ? g[j]  : 0.0f;
    lds[(256 + tid) * 16 + j] = (j < 6) ? xs[j] : 0.0f;
  }
  __syncthreads();

  // ---- Gram reduction via V_WMMA_F32_16X16X4_F32 ----
  // Per WMMA: A[m][k] = v_{e0+k}[m] (16x4), B[k][n] = v_{e0+k}[n] (4x16),
  // D += A*B accumulates sum_k v v^T. xi* rows (>=256) get A negated -> subtract.
  const int w    = tid >> 5;         // wave id 0..7
  const int lane = tid & 31;
  const int half = lane >> 4;        // 0: lanes 0-15, 1: lanes 16-31
  const int lm   = lane & 15;
  const float sgn = (w >= 4) ? -1.0f : 1.0f;
  const int base  = w * 64;          // each wave owns 64 of the 512 rows

  v8f acc = {};
  #pragma unroll
  for (int i = 0; i < 16; ++i) {
    int e0 = base + i * 4;
    v2f a, bb;
    // A (16x4 f32): vgpr0 lanes0-15 K=0 | lanes16-31 K=2 ; vgpr1: K=1 | K=3
    a.x  = lds[(e0 + 0 + 2*half) * 16 + lm] * sgn;
    a.y  = lds[(e0 + 1 + 2*half) * 16 + lm] * sgn;
    // B (4x16 f32): vgpr0 lanes0-15 K=0 row | lanes16-31 K=1 ; vgpr1: K=2 | K=3
    bb.x = lds[(e0 + 0 + half) * 16 + lm];
    bb.y = lds[(e0 + 2 + half) * 16 + lm];
    acc = __builtin_amdgcn_wmma_f32_16x16x4_f32(
        /*neg_a=*/false, a, /*neg_b=*/false, bb,
        /*c_mod=*/(short)0, acc, /*reuse_a=*/false, /*reuse_b=*/false);
  }

  // Contract D tile with precision: D(VGPR j, lane) = (M = j + 8*half, N = lm).
  // Only M<6, N<6 is nonzero -> lanes with half==1 (M>=8) contribute nothing.
  float s = 0.0f;
  if (half == 0 && lm < 6) {
    #pragma unroll
    for (int j = 0; j < 6; ++j)
      s += acc[j] * precision[j * 6 + lm];
  }

  red[tid] = s;
  __syncthreads();
  #pragma unroll
  for (int off = 128; off > 0; off >>= 1) {
    if (tid < off) red[tid] += red[tid + off];
    __syncthreads();
  }
  if (tid == 0) partials[blockIdx.x] = red[0];
}

__global__ __launch_bounds__(256) void se3_loss_final(
    const float* __restrict__ partials, int nblocks,
    float* __restrict__ out, float scale)
{
  __shared__ float red[256];
  const int tid = threadIdx.x;
  float s = 0.0f;
  for (int i = tid; i < nblocks; i += 256) s += partials[i];
  red[tid] = s;
  __syncthreads();
  #pragma unroll
  for (int off = 128; off > 0; off >>= 1) {
    if (tid < off) red[tid] += red[tid + off];
    __syncthreads();
  }
  if (tid == 0) out[0] = red[0] * scale;
}

extern "C" void kernel_launch(void* const* d_in, const int* in_sizes, int n_in,
                              void* d_out, int out_size, void* d_ws, size_t ws_size,
                              hipStream_t stream) {
  const float* input     = (const float*)d_in[0];  // (B,6)
  const float* target    = (const float*)d_in[1];  // (B,4,4)
  const float* precision = (const float*)d_in[2];  // (6,6)
  float* out      = (float*)d_out;
  float* partials = (float*)d_ws;

  const int B       = in_sizes[0] / 6;
  const int nblocks = (B + 255) / 256;

  se3_loss_main<<<nblocks, 256, 0, stream>>>(input, target, precision, partials, B);
  se3_loss_final<<<1, 256, 0, stream>>>(partials, nblocks, out, 0.5f / (float)B);
}